// BaseTBLoRa_76794015252534
// MI455X (gfx1250) — compile-verified
//
#include <hip/hip_runtime.h>
#include <hip/hip_bf16.h>

typedef __attribute__((ext_vector_type(2))) float v2f;
typedef __attribute__((ext_vector_type(4))) float f4;
typedef __attribute__((ext_vector_type(8))) float v8f;
typedef __attribute__((ext_vector_type(4))) int   v4i;

#define D_IN   1024
#define D_OUT  1024
#define SEQ    2048
#define RANK   16
#define BM     128
#define BN     128
#define KC     16
#define LDK    20      // padded LDS row stride (floats); mult of 4 -> 16B aligned
#define NCHUNK 64      // D_IN / KC
#define LORA_SCALE 2.0f

// ---- CDNA5 async global->LDS path (guarded; falls back to sync copy) ------
#if __has_builtin(__builtin_amdgcn_global_load_async_to_lds_b128)
#define HAVE_ASYNC_LDS 1
__device__ __forceinline__ void cp16(const float* g, float* l) {
    __builtin_amdgcn_global_load_async_to_lds_b128((v4i*)g, (v4i*)l, 0, 0);
}
#if __has_builtin(__builtin_amdgcn_s_wait_asynccnt)
#define WAIT_ASYNC(N) __builtin_amdgcn_s_wait_asynccnt(N)
#else
#define WAIT_ASYNC(N) asm volatile("s_wait_asynccnt %0" ::"i"(N) : "memory")
#endif
#else
#define HAVE_ASYNC_LDS 0
__device__ __forceinline__ void cp16(const float* g, float* l) {
    *(f4*)l = *(const f4*)g;           // synchronous fallback
}
#define WAIT_ASYNC(N) do { } while (0)
#endif

// ---------------------------------------------------------------------------
// Pass 1: H[m, r] = LORA_SCALE * sum_d x[m, d] * A_w[seg(m), r, d]
// One wave computes one 16(M) x 16(R) tile via V_WMMA_F32_16X16X4_F32.
// ---------------------------------------------------------------------------
__global__ __launch_bounds__(256) void lora_h_kernel(
    const float* __restrict__ x, const int* __restrict__ segment,
    const float* __restrict__ A_w, float* __restrict__ H)
{
    const int lane = threadIdx.x & 31;
    const int wid  = threadIdx.x >> 5;
    const int l15  = lane & 15;
    const int half = lane >> 4;

    const int m0    = blockIdx.x * 128 + wid * 16;   // 16 rows per wave
    const int batch = m0 >> 11;                      // / SEQ
    const int seg   = segment[batch];

    const float* xrow = x   + (size_t)(m0 + l15) * D_IN + 2 * half;
    const float* arow = A_w + ((size_t)seg * RANK + l15) * D_IN + 2 * half;

    v8f acc = {};
    for (int k0 = 0; k0 < D_IN; k0 += 4) {
        v2f a = *(const v2f*)(xrow + k0);
        v2f b = *(const v2f*)(arow + k0);
        acc = __builtin_amdgcn_wmma_f32_16x16x4_f32(
            false, a, false, b, (short)0, acc, false, false);
    }

#pragma unroll
    for (int r = 0; r < 8; ++r) {
        const int m = m0 + r + 8 * half;
        H[(size_t)m * RANK + l15] = acc[r] * LORA_SCALE;
    }
}

// ---------------------------------------------------------------------------
// One K-chunk (KC=16) of WMMA work out of LDS. Wave tile: 64(M) x 32(N).
// ---------------------------------------------------------------------------
__device__ __forceinline__ void mma_chunk(const float* __restrict__ Xs,
                                          const float* __restrict__ Ws,
                                          int wm, int wn, int l15, int half,
                                          v8f acc[4][2])
{
#pragma unroll
    for (int ks = 0; ks < 4; ++ks) {
        const int k0 = ks * 4 + 2 * half;
        v2f a[4], b[2];
#pragma unroll
        for (int i = 0; i < 4; ++i)
            a[i] = *(const v2f*)(&Xs[(wm + i * 16 + l15) * LDK + k0]);
#pragma unroll
        for (int j = 0; j < 2; ++j)
            b[j] = *(const v2f*)(&Ws[(wn + j * 16 + l15) * LDK + k0]);
#pragma unroll
        for (int i = 0; i < 4; ++i)
#pragma unroll
            for (int j = 0; j < 2; ++j)
                acc[i][j] = __builtin_amdgcn_wmma_f32_16x16x4_f32(
                    false, a[i], false, b[j], (short)0, acc[i][j], false, false);
    }
}

// ---------------------------------------------------------------------------
// Pass 2: out = X*W^T + bias + H*B_w[seg]^T  (LoRA term = chunk #NCHUNK)
// 128x128 block tile, 8 waves (2x4), double-buffered LDS filled with
// GLOBAL_LOAD_ASYNC_TO_LDS_B128, synchronized via s_wait_asynccnt.
// ---------------------------------------------------------------------------
__global__ __launch_bounds__(256) void lora_main_kernel(
    const float* __restrict__ x, const int* __restrict__ segment,
    const float* __restrict__ W, const float* __restrict__ bias,
    const float* __restrict__ Bw, const float* __restrict__ H,
    float* __restrict__ out)
{
    __shared__ float Xs[2 * BM * LDK];
    __shared__ float Ws[2 * BN * LDK];

    const int tid  = threadIdx.x;
    const int lane = tid & 31;
    const int wid  = tid >> 5;
    const int l15  = lane & 15;
    const int half = lane >> 4;

    const int gm0 = blockIdx.x * BM;
    const int gn0 = blockIdx.y * BN;
    const int wm  = (wid >> 2) * 64;   // 0 or 64
    const int wn  = (wid & 3) * 32;    // 0,32,64,96

    const int lrow = tid >> 1;         // 0..127
    const int lcol = (tid & 1) * 8;    // 0 or 8
    const int seg  = segment[gm0 >> 11];
    const int loff = lrow * LDK + lcol;

    v8f acc[4][2];
    {
        v8f z = {};
#pragma unroll
        for (int i = 0; i < 4; ++i)
#pragma unroll
            for (int j = 0; j < 2; ++j) acc[i][j] = z;
    }

    // issue this thread's 4 x b128 slice of chunk c into (dX, dW)
    auto issue_chunk = [&](int c, float* dX, float* dW) {
        const float *gx, *gw;
        if (c < NCHUNK) {
            gx = x + (size_t)(gm0 + lrow) * D_IN + c * KC + lcol;
            gw = W + (size_t)(gn0 + lrow) * D_IN + c * KC + lcol;
        } else {  // LoRA chunk: H (pre-scaled) and B_w[seg]
            gx = H  + (size_t)(gm0 + lrow) * RANK + lcol;
            gw = Bw + ((size_t)seg * D_OUT + gn0 + lrow) * RANK + lcol;
        }
        cp16(gx,     dX);
        cp16(gx + 4, dX + 4);
        cp16(gw,     dW);
        cp16(gw + 4, dW + 4);
    };

    // prologue: start chunk 0 into buffer 0
    issue_chunk(0, Xs + loff, Ws + loff);

    for (int c = 0; c <= NCHUNK; ++c) {
        float* curX = Xs + (c & 1) * (BM * LDK);
        float* curW = Ws + (c & 1) * (BN * LDK);
        float* nxtX = Xs + ((c + 1) & 1) * (BM * LDK);
        float* nxtW = Ws + ((c + 1) & 1) * (BN * LDK);

        __syncthreads();                       // readers of `nxt` (iter c-1) done
        if (c < NCHUNK) {
            issue_chunk(c + 1, nxtX + loff, nxtW + loff);
            WAIT_ASYNC(4);                     // chunk c landed; c+1 still in flight
        } else {
            WAIT_ASYNC(0);                     // drain final (LoRA) chunk
        }
        __syncthreads();                       // chunk c visible to all waves

        mma_chunk(curX, curW, wm, wn, l15, half, acc);
    }

    // ---- epilogue: + bias, store ----
#pragma unroll
    for (int j = 0; j < 2; ++j) {
        const int n   = gn0 + wn + j * 16 + l15;
        const float bb = bias[n];
#pragma unroll
        for (int i = 0; i < 4; ++i) {
#pragma unroll
            for (int r = 0; r < 8; ++r) {
                const int m = gm0 + wm + i * 16 + r + 8 * half;
                out[(size_t)m * D_OUT + n] = acc[i][j][r] + bb;
            }
        }
    }
}

// ---------------------------------------------------------------------------
extern "C" void kernel_launch(void* const* d_in, const int* in_sizes, int n_in,
                              void* d_out, int out_size, void* d_ws, size_t ws_size,
                              hipStream_t stream) {
    const float* x       = (const float*)d_in[0];   // [8,2048,1024]
    const int*   segment = (const int*)d_in[1];     // [8]
    const float* W_base  = (const float*)d_in[2];   // [1024,1024]
    const float* b_base  = (const float*)d_in[3];   // [1024]
    const float* A_w     = (const float*)d_in[4];   // [8,16,1024]
    const float* B_w     = (const float*)d_in[5];   // [8,1024,16]
    float*       out     = (float*)d_out;           // [8,2048,1024]
    float*       H       = (float*)d_ws;            // [16384,16] scratch (1 MB)

    const int M = 8 * SEQ;                          // 16384

    lora_h_kernel<<<M / 128, 256, 0, stream>>>(x, segment, A_w, H);

    dim3 grid(M / BM, D_OUT / BN);                  // 128 x 8
    lora_main_kernel<<<grid, 256, 0, stream>>>(x, segment, W_base, b_base,
                                               B_w, H, out);
}